// AdaptivePConv_88854283420299
// MI455X (gfx1250) — compile-verified
//
#include <hip/hip_runtime.h>
#include <hip/hip_bf16.h>

typedef float v2f __attribute__((ext_vector_type(2)));
typedef float v8f __attribute__((ext_vector_type(8)));

#define Bn   32
#define Cn   256
#define Hn   128
#define Wn   128
#define Cp   64
#define Co   256
#define HW   (Hn*Wn)          // 16384
#define Cout 448              // 256 + 192
#define Cun  192

#define WT   32               // output W-tile per block
#define LCOL (WT + 2)         // 34 input columns staged
// LDS float layout: [kh][cipair][col][2]  (channel pairs interleaved)
#define LDS_IDX(kh, cp, col) ((((kh)*32 + (cp))*LCOL + (col))*2)

// ---------------------------------------------------------------------------
// 1) per-(b,c) sum / sumsq -> std (ddof=1)
// ---------------------------------------------------------------------------
__global__ __launch_bounds__(256)
void AP_stats_kernel(const float* __restrict__ x, float* __restrict__ stdv) {
    const float4* p = (const float4*)(x + (size_t)blockIdx.x * HW);
    float s = 0.f, ss = 0.f;
    for (int i = threadIdx.x; i < HW/4; i += 256) {
        float4 v = p[i];
        s  += v.x + v.y + v.z + v.w;
        ss += v.x*v.x + v.y*v.y + v.z*v.z + v.w*v.w;
    }
    __shared__ float sh_s[256], sh_q[256];
    sh_s[threadIdx.x] = s; sh_q[threadIdx.x] = ss;
    __syncthreads();
    for (int off = 128; off > 0; off >>= 1) {
        if (threadIdx.x < off) {
            sh_s[threadIdx.x] += sh_s[threadIdx.x + off];
            sh_q[threadIdx.x] += sh_q[threadIdx.x + off];
        }
        __syncthreads();
    }
    if (threadIdx.x == 0) {
        const float N = (float)HW;
        float mean = sh_s[0] / N;
        float var  = (sh_q[0] - sh_s[0]*mean) / (N - 1.0f);
        stdv[blockIdx.x] = sqrtf(var > 0.f ? var : 0.f);
    }
}

// ---------------------------------------------------------------------------
// 2) weights (Co,Cp,3,3) -> Wt2[tap][ci/2][co][2]   (ci pairs interleaved)
// ---------------------------------------------------------------------------
__global__ __launch_bounds__(256)
void AP_wtrans_kernel(const float* __restrict__ Wconv, float* __restrict__ Wt2) {
    int gid = blockIdx.x * 256 + threadIdx.x;      // 576*256 == Co*Cp*9
    int co  = gid / (Cp*9);
    int rem = gid % (Cp*9);
    int ci  = rem / 9;
    int tap = rem % 9;
    Wt2[(((size_t)tap*(Cp/2) + (ci>>1))*Co + co)*2 + (ci & 1)] = Wconv[gid];
}

// ---------------------------------------------------------------------------
// 3) top-64 per batch (desc value, tie -> lower index) + unsel list (asc index)
// ---------------------------------------------------------------------------
__global__ __launch_bounds__(256)
void AP_topk_kernel(const float* __restrict__ stdv,
                    int* __restrict__ top_idx, int* __restrict__ unsel_idx) {
    const int b = blockIdx.x, c = threadIdx.x;
    __shared__ float v[Cn];
    __shared__ int   flag[Cn];
    v[c] = stdv[b*Cn + c];
    __syncthreads();
    float mv = v[c];
    int rank = 0;
    for (int j = 0; j < Cn; ++j) {
        float o = v[j];
        rank += (o > mv) || (o == mv && j < c);
    }
    int f = (rank < Cp);
    flag[c] = f;
    if (f) top_idx[b*Cp + rank] = c;
    __syncthreads();
    if (!f) {
        int pos = 0;
        for (int j = 0; j < c; ++j) pos += (flag[j] == 0);
        unsel_idx[b*Cun + pos] = c;
    }
}

// ---------------------------------------------------------------------------
// 4) implicit-GEMM 3x3 conv with V_WMMA_F32_16X16X4_F32
//    M = 16 output channels, N = 16 W positions, K = input channel (x9 taps)
//    grid (Wn/32, Hn, Bn), 256 threads = 8 waves
//    each wave: 2 co-tiles x 2 w-tiles (4 accumulators), so per k-step
//    2x global_load_b64 (A) + 2x ds_load_b64 (B) feed 4 WMMAs.
// ---------------------------------------------------------------------------
__global__ __launch_bounds__(256)
void AP_conv_wmma_kernel(const float* __restrict__ x,
                         const float* __restrict__ Wt2,
                         const float* __restrict__ bconv,
                         const int*   __restrict__ top_idx,
                         float* __restrict__ out) {
    __shared__ float lds[3 * 32 * LCOL * 2];       // 26112 B
    const int w0  = blockIdx.x * WT;
    const int h   = blockIdx.y;
    const int b   = blockIdx.z;
    const int tid = threadIdx.x;

    const int* tsel = top_idx + b * Cp;
    for (int idx = tid; idx < Cp*3*LCOL; idx += 256) {
        int ci  = idx / (3*LCOL);
        int rem = idx % (3*LCOL);
        int kh  = rem / LCOL;
        int col = rem % LCOL;
        int hs  = h + kh - 1;
        int ws  = w0 + col - 1;
        float val = 0.f;
        if (hs >= 0 && hs < Hn && ws >= 0 && ws < Wn) {
            int ch = tsel[ci];
            val = x[(((size_t)b*Cn + ch)*Hn + hs)*Wn + ws];
        }
        lds[LDS_IDX(kh, ci >> 1, col) + (ci & 1)] = val;
    }
    __syncthreads();

    const int wid  = tid >> 5;
    const int lane = tid & 31;
    const int half = lane >> 4;        // 0: lanes 0-15 hold K0/K1, M+0 ; 1: K2/K3, M+8
    const int lm   = lane & 15;
    const int co0  = wid * 32;         // wave covers output channels [co0, co0+32)
    const int co1  = co0 + 16;

    v8f acc00 = {}, acc01 = {}, acc10 = {}, acc11 = {};
    for (int tap = 0; tap < 9; ++tap) {
        const int kh = tap / 3, kw = tap % 3;
        const float* wt = Wt2 + (size_t)tap * (Cp/2) * Co * 2;
        const float* lrow = lds + (size_t)kh * 32 * LCOL * 2;
        for (int ci0 = 0; ci0 < Cp; ci0 += 4) {
            const int cp = (ci0 >> 1) + half;      // channel-pair index for this lane half
            // A fragments (weights, M x K): one b64 per co-tile
            const float* wr = wt + ((size_t)cp * Co) * 2;
            v2f A0 = *(const v2f*)(wr + (co0 + lm)*2);
            v2f A1 = *(const v2f*)(wr + (co1 + lm)*2);
            // B fragments (input patches, K x N): one ds b64 per w-tile
            const float* br = lrow + ((size_t)cp * LCOL)*2;
            v2f B0 = *(const v2f*)(br + (lm + kw)*2);
            v2f B1 = *(const v2f*)(br + (16 + lm + kw)*2);
            acc00 = __builtin_amdgcn_wmma_f32_16x16x4_f32(
                        false, A0, false, B0, (short)0, acc00, false, false);
            acc01 = __builtin_amdgcn_wmma_f32_16x16x4_f32(
                        false, A0, false, B1, (short)0, acc01, false, false);
            acc10 = __builtin_amdgcn_wmma_f32_16x16x4_f32(
                        false, A1, false, B0, (short)0, acc10, false, false);
            acc11 = __builtin_amdgcn_wmma_f32_16x16x4_f32(
                        false, A1, false, B1, (short)0, acc11, false, false);
        }
    }

    // D layout: VGPR r -> M = r + 8*half, N = lm (coalesced 64B runs per half)
    for (int r = 0; r < 8; ++r) {
        int m   = r + half*8;
        int coA = co0 + m;
        int coB = co1 + m;
        float bA = bconv[coA];
        float bB = bconv[coB];
        size_t baseA = (((size_t)b*Cout + coA)*Hn + h)*Wn + w0 + lm;
        size_t baseB = (((size_t)b*Cout + coB)*Hn + h)*Wn + w0 + lm;
        out[baseA]      = acc00[r] + bA;
        out[baseA + 16] = acc01[r] + bA;
        out[baseB]      = acc10[r] + bB;
        out[baseB + 16] = acc11[r] + bB;
    }
}

// ---------------------------------------------------------------------------
// 5) copy 192 unselected channels to out[:, 256:448]
// ---------------------------------------------------------------------------
__global__ __launch_bounds__(256)
void AP_unsel_kernel(const float* __restrict__ x,
                     const int* __restrict__ unsel_idx,
                     float* __restrict__ out) {
    const int j = blockIdx.x;          // 0..191
    const int b = blockIdx.y;
    const int ch = unsel_idx[b*Cun + j];
    const float4* src = (const float4*)(x   + ((size_t)b*Cn   + ch     ) * HW);
    float4*       dst = (float4*)      (out + ((size_t)b*Cout + Cn + j ) * HW);
    for (int i = threadIdx.x; i < HW/4; i += 256) dst[i] = src[i];
}

// ---------------------------------------------------------------------------
extern "C" void kernel_launch(void* const* d_in, const int* in_sizes, int n_in,
                              void* d_out, int out_size, void* d_ws, size_t ws_size,
                              hipStream_t stream) {
    const float* x     = (const float*)d_in[0];   // (32,256,128,128)
    const float* Wconv = (const float*)d_in[1];   // (256,64,3,3)
    const float* bconv = (const float*)d_in[2];   // (256,)
    float* out = (float*)d_out;                   // (32,448,128,128)

    char* ws = (char*)d_ws;
    float* stdv      = (float*)(ws);                         // 8192 f
    int*   top_idx   = (int*)  (ws + 32768);                 // 32*64 i32
    int*   unsel_idx = (int*)  (ws + 40960);                 // 32*192 i32
    float* Wt2       = (float*)(ws + 65536);                 // 9*64*256 f (576KB)

    AP_stats_kernel <<<Bn*Cn, 256, 0, stream>>>(x, stdv);
    AP_wtrans_kernel<<<(Co*Cp*9)/256, 256, 0, stream>>>(Wconv, Wt2);
    AP_topk_kernel  <<<Bn, 256, 0, stream>>>(stdv, top_idx, unsel_idx);
    AP_conv_wmma_kernel<<<dim3(Wn/WT, Hn, Bn), 256, 0, stream>>>(
        x, Wt2, bconv, top_idx, out);
    AP_unsel_kernel <<<dim3(Cun, Bn), 256, 0, stream>>>(x, unsel_idx, out);
}